// ToxicityGAT_52561809768558
// MI455X (gfx1250) — compile-verified
//
#include <hip/hip_runtime.h>
#include <hip/hip_bf16.h>
#include <math.h>

typedef __attribute__((ext_vector_type(2))) float v2f;
typedef __attribute__((ext_vector_type(8))) float v8f;

#define TPB 256

// ---------------------------------------------------------------------------
// fp32 WMMA GEMM: C[M,N] = A[M,K] @ B[K,N], all row-major.
// One wave32 computes a 16 x (16*NT) strip of C via V_WMMA_F32_16X16X4_F32,
// reusing the A fragment across NT column tiles each K-step.
// Requires M%16==0, N%(16*NT)==0, K%4==0 (M=50000=3125*16, N in {256,64}).
//
// ISA lane layouts (CDNA5 ISA):
//   A 16x4 : lane L -> m=L%16, vgpr{0,1} = K = 2*(L/16)+{0,1}
//   B 4x16 : lane L -> n=L%16, vgpr{0,1} = K = 2*(L/16)+{0,1}   (mirror of A)
//   C 16x16: lane L, vgpr j -> m = j + 8*(L/16), n = L%16
// ---------------------------------------------------------------------------
template <int NT>
__global__ void gat_gemm_wmma_f32(const float* __restrict__ A,
                                  const float* __restrict__ B,
                                  float* __restrict__ C,
                                  int M, int Ncols, int K) {
  const int lane = threadIdx.x & 31;
  const int wave = threadIdx.x >> 5;
  const int wavesPerBlock = blockDim.x >> 5;
  const int groupsN = Ncols / (16 * NT);
  const int totalGroups = (M >> 4) * groupsN;
  const int gid = blockIdx.x * wavesPerBlock + wave;
  if (gid >= totalGroups) return;

  const int tm = gid / groupsN;
  const int tg = gid - tm * groupsN;
  const int row0 = tm << 4;
  const int col0 = tg * (16 * NT);
  const int half = lane >> 4;   // 0 or 1
  const int l16  = lane & 15;

  const float* __restrict__ Arow  = A + (size_t)(row0 + l16) * K + 2 * half;
  const float* __restrict__ Bbase = B + (size_t)(2 * half) * Ncols + col0 + l16;

  v8f acc[NT];
#pragma unroll
  for (int t = 0; t < NT; ++t) acc[t] = (v8f){};

  for (int k = 0; k < K; k += 4) {
    v2f a;
    a.x = Arow[k + 0];
    a.y = Arow[k + 1];
    const float* __restrict__ Bk = Bbase + (size_t)k * Ncols;
#pragma unroll
    for (int t = 0; t < NT; ++t) {
      v2f b;
      b.x = Bk[t * 16];
      b.y = Bk[(size_t)Ncols + t * 16];
      acc[t] = __builtin_amdgcn_wmma_f32_16x16x4_f32(
          /*neg_a=*/false, a, /*neg_b=*/false, b,
          /*c_mod=*/(short)0, acc[t], /*reuse_a=*/false, /*reuse_b=*/false);
    }
  }

  float* __restrict__ Crow = C + (size_t)(row0 + 8 * half) * Ncols + col0 + l16;
#pragma unroll
  for (int t = 0; t < NT; ++t)
#pragma unroll
    for (int j = 0; j < 8; ++j)
      Crow[(size_t)j * Ncols + t * 16] = acc[t][j];
}

// ---------------------------------------------------------------------------
// helpers
// ---------------------------------------------------------------------------
__device__ __forceinline__ float leaky02(float x) {
  return x > 0.0f ? x : 0.2f * x;
}
__device__ __forceinline__ float elu1(float x) {
  return x > 0.0f ? x : (expf(x) - 1.0f);
}
// Monotone float atomic-max via int/uint punning (init value must be -inf).
__device__ __forceinline__ void atomicMaxF32(float* addr, float val) {
  if (val >= 0.0f) atomicMax((int*)addr, __float_as_int(val));
  else             atomicMin((unsigned int*)addr, (unsigned int)__float_as_int(val));
}
__device__ __forceinline__ void edgeEndpoints(const long long* __restrict__ ei,
                                              int E, long long e, int& s, int& d) {
  if (e < (long long)E) { s = (int)ei[e]; d = (int)ei[(size_t)E + e]; }
  else                  { s = d = (int)(e - E); }   // self-loop appended
}

__global__ void gat_fill_f32(float* __restrict__ p, float v, long long n) {
  long long i = (long long)blockIdx.x * blockDim.x + threadIdx.x;
  if (i < n) p[i] = v;
}

// alpha_s[n,h] = dot(hpre[n, h*D:(h+1)*D], att_src[h]); same for alpha_d.
__global__ void gat_scores(const float* __restrict__ hpre,
                           const float* __restrict__ att_s,
                           const float* __restrict__ att_d,
                           float* __restrict__ as, float* __restrict__ ad,
                           int n, int H, int D) {
  int i = blockIdx.x * blockDim.x + threadIdx.x;
  if (i >= n * H) return;
  int node = i / H, h = i - node * H;
  const float* __restrict__ row = hpre + (size_t)node * H * D + (size_t)h * D;
  const float* __restrict__ s = att_s + (size_t)h * D;
  const float* __restrict__ d = att_d + (size_t)h * D;
  float accs = 0.0f, accd = 0.0f;
  for (int k = 0; k < D; ++k) { float v = row[k]; accs += v * s[k]; accd += v * d[k]; }
  as[i] = accs;
  ad[i] = accd;
}

// segment-max of leaky(a_s[src]+a_d[dst]) into m[dst,h]
__global__ void gat_edge_max(const long long* __restrict__ ei, int E, int n,
                             const float* __restrict__ as, const float* __restrict__ ad,
                             float* __restrict__ m, int H) {
  long long e = (long long)blockIdx.x * blockDim.x + threadIdx.x;
  if (e >= (long long)E + n) return;
  int s, d; edgeEndpoints(ei, E, e, s, d);
  for (int h = 0; h < H; ++h) {
    float v = leaky02(as[(size_t)s * H + h] + ad[(size_t)d * H + h]);
    atomicMaxF32(&m[(size_t)d * H + h], v);
  }
}

// ex[e,h] = exp(e - m[dst]);  den[dst,h] += ex
__global__ void gat_edge_exp(const long long* __restrict__ ei, int E, int n,
                             const float* __restrict__ as, const float* __restrict__ ad,
                             const float* __restrict__ m,
                             float* __restrict__ ex, float* __restrict__ den, int H) {
  long long e = (long long)blockIdx.x * blockDim.x + threadIdx.x;
  if (e >= (long long)E + n) return;
  int s, d; edgeEndpoints(ei, E, e, s, d);
  for (int h = 0; h < H; ++h) {
    float v = leaky02(as[(size_t)s * H + h] + ad[(size_t)d * H + h]);
    float xv = expf(v - m[(size_t)d * H + h]);
    ex[(size_t)e * H + h] = xv;
    atomicAdd(&den[(size_t)d * H + h], xv);
  }
}

// agg[dst, c] += hpre[src, c] * ex[e, h]/den[dst, h]
// one block per edge, one thread per channel c in [0, H*D)  (coalesced)
__global__ void gat_edge_scatter(const long long* __restrict__ ei, int E, int n,
                                 const float* __restrict__ hpre,
                                 const float* __restrict__ ex,
                                 const float* __restrict__ den,
                                 float* __restrict__ agg, int H, int D) {
  long long e = blockIdx.x;
  int c = threadIdx.x;
  int HD = H * D;
  if (c >= HD || e >= (long long)E + n) return;
  int s, d; edgeEndpoints(ei, E, e, s, d);
  int h = c / D;
  float a = ex[(size_t)e * H + h] / den[(size_t)d * H + h];
  atomicAdd(&agg[(size_t)d * HD + c], hpre[(size_t)s * HD + c] * a);
}

// out[i] = elu(agg[i] + b[i % C]); C is a power of two (256 or 64)
__global__ void gat_bias_elu(float* __restrict__ agg, const float* __restrict__ b,
                             long long total, int Cmask) {
  long long i = (long long)blockIdx.x * blockDim.x + threadIdx.x;
  if (i >= total) return;
  agg[i] = elu1(agg[i] + b[(int)(i & Cmask)]);
}

// out[n, 0:2] = h2[n, :] @ Wc + bc
__global__ void gat_classify(const float* __restrict__ h2, const float* __restrict__ Wc,
                             const float* __restrict__ bc, float* __restrict__ out, int n) {
  int node = blockIdx.x * blockDim.x + threadIdx.x;
  if (node >= n) return;
  const float* __restrict__ r = h2 + (size_t)node * 64;
  float a0 = bc[0], a1 = bc[1];
  for (int k = 0; k < 64; ++k) {
    float v = r[k];
    a0 += v * Wc[2 * k + 0];
    a1 += v * Wc[2 * k + 1];
  }
  out[(size_t)node * 2 + 0] = a0;
  out[(size_t)node * 2 + 1] = a1;
}

// ---------------------------------------------------------------------------
extern "C" void kernel_launch(void* const* d_in, const int* in_sizes, int n_in,
                              void* d_out, int out_size, void* d_ws, size_t ws_size,
                              hipStream_t stream) {
  const float*     x        = (const float*)d_in[0];
  const long long* ei       = (const long long*)d_in[1];   // int64 edge_index [2,E]
  const float*     W1       = (const float*)d_in[2];
  const float*     att1_src = (const float*)d_in[3];
  const float*     att1_dst = (const float*)d_in[4];
  const float*     b1       = (const float*)d_in[5];
  const float*     W2       = (const float*)d_in[6];
  const float*     att2_src = (const float*)d_in[7];
  const float*     att2_dst = (const float*)d_in[8];
  const float*     b2       = (const float*)d_in[9];
  const float*     Wc       = (const float*)d_in[10];
  const float*     bc       = (const float*)d_in[11];
  float*           out      = (float*)d_out;

  const int IN_DIM = 128, HID = 64, HEADS = 4;
  const int N = in_sizes[0] / IN_DIM;       // 50000
  const int E = in_sizes[1] / 2;            // 800000
  const long long E2 = (long long)E + N;    // with self-loops

  // ---- workspace layout (floats) ----
  float* w = (float*)d_ws;
  size_t o = 0;
  float* h1_pre = w + o; o += (size_t)N * 256;   // x@W1 ; later reused: h2_pre + agg2
  float* agg1   = w + o; o += (size_t)N * 256;   // scatter target -> h1 (in place)
  float* ex     = w + o; o += (size_t)E2 * 4;    // edge exp; reused for layer 2 (E2*1)
  float* as1    = w + o; o += (size_t)N * 4;
  float* ad1    = w + o; o += (size_t)N * 4;
  float* m1     = w + o; o += (size_t)N * 4;
  float* den1   = w + o; o += (size_t)N * 4;
  float* as2    = w + o; o += (size_t)N;
  float* ad2    = w + o; o += (size_t)N;
  float* m2     = w + o; o += (size_t)N;
  float* den2   = w + o; o += (size_t)N;
  float* h2_pre = h1_pre;                        // N*64, reuse (h1_pre dead after scatter1)
  float* agg2   = h1_pre + (size_t)N * 64;       // N*64, still inside h1_pre region

  auto blocks = [](long long total, int tpb) { return (unsigned)((total + tpb - 1) / tpb); };

  // ================= Layer 1: GATConv(128 -> 4 heads x 64, concat) =========
  {
    int groups = (N / 16) * (256 / 64);   // 16x64 strips per wave
    gat_gemm_wmma_f32<4><<<blocks(groups, 8), TPB, 0, stream>>>(x, W1, h1_pre, N, 256, IN_DIM);
  }
  gat_scores<<<blocks((long long)N * HEADS, TPB), TPB, 0, stream>>>(
      h1_pre, att1_src, att1_dst, as1, ad1, N, HEADS, HID);

  gat_fill_f32<<<blocks((long long)N * HEADS, TPB), TPB, 0, stream>>>(m1, -INFINITY, (long long)N * HEADS);
  gat_fill_f32<<<blocks((long long)N * HEADS, TPB), TPB, 0, stream>>>(den1, 0.0f, (long long)N * HEADS);
  gat_fill_f32<<<blocks((long long)N * 256, TPB), TPB, 0, stream>>>(agg1, 0.0f, (long long)N * 256);

  gat_edge_max<<<blocks(E2, TPB), TPB, 0, stream>>>(ei, E, N, as1, ad1, m1, HEADS);
  gat_edge_exp<<<blocks(E2, TPB), TPB, 0, stream>>>(ei, E, N, as1, ad1, m1, ex, den1, HEADS);
  gat_edge_scatter<<<(unsigned)E2, 256, 0, stream>>>(ei, E, N, h1_pre, ex, den1, agg1, HEADS, HID);
  gat_bias_elu<<<blocks((long long)N * 256, TPB), TPB, 0, stream>>>(agg1, b1, (long long)N * 256, 255);
  // agg1 now holds h1 [N,256]

  // ================= Layer 2: GATConv(256 -> 1 head x 64) ==================
  {
    int groups = (N / 16) * (64 / 64);
    gat_gemm_wmma_f32<4><<<blocks(groups, 8), TPB, 0, stream>>>(agg1, W2, h2_pre, N, 64, 256);
  }
  gat_scores<<<blocks((long long)N, TPB), TPB, 0, stream>>>(
      h2_pre, att2_src, att2_dst, as2, ad2, N, 1, HID);

  gat_fill_f32<<<blocks((long long)N, TPB), TPB, 0, stream>>>(m2, -INFINITY, (long long)N);
  gat_fill_f32<<<blocks((long long)N, TPB), TPB, 0, stream>>>(den2, 0.0f, (long long)N);
  gat_fill_f32<<<blocks((long long)N * 64, TPB), TPB, 0, stream>>>(agg2, 0.0f, (long long)N * 64);

  gat_edge_max<<<blocks(E2, TPB), TPB, 0, stream>>>(ei, E, N, as2, ad2, m2, 1);
  gat_edge_exp<<<blocks(E2, TPB), TPB, 0, stream>>>(ei, E, N, as2, ad2, m2, ex, den2, 1);
  gat_edge_scatter<<<(unsigned)E2, 64, 0, stream>>>(ei, E, N, h2_pre, ex, den2, agg2, 1, HID);
  gat_bias_elu<<<blocks((long long)N * 64, TPB), TPB, 0, stream>>>(agg2, b2, (long long)N * 64, 63);
  // agg2 now holds h2 [N,64]

  // ================= classifier =============================================
  gat_classify<<<blocks((long long)N, TPB), TPB, 0, stream>>>(agg2, Wc, bc, out, N);
}